// CrossNet_80719615361476
// MI455X (gfx1250) — compile-verified
//
#include <hip/hip_runtime.h>

// CDNA5 / gfx1250 — wave32, WMMA f32 16x16x4.
typedef __attribute__((ext_vector_type(2))) float v2f;
typedef __attribute__((ext_vector_type(8))) float v8f;

#define DIM 1024
#define WPAD 1028                     // padded LDS row stride (banks: 4*wIdx offset)
#define ROWS_PER_WAVE 16
#define WAVES_PER_BLOCK 4
#define ROWS_PER_BLOCK (ROWS_PER_WAVE * WAVES_PER_BLOCK)

__global__ __launch_bounds__(128) void crossnet_wmma_kernel(
    const float* __restrict__ x0,   // [16384, 1024]
    const float* __restrict__ W,    // [3, 1024]
    const float* __restrict__ B,    // [3, 1024]
    float* __restrict__ out)        // [16384, 1024]
{
    __shared__ __align__(16) float WS[4 * WPAD];      // W rows 0..2 + zero row 3
    __shared__ __align__(16) float BsumS[DIM];        // B0+B1+B2
    __shared__ float dts[WAVES_PER_BLOCK][3][16];     // per-wave D columns 0..2
    __shared__ float alphaS[WAVES_PER_BLOCK][16];     // per-row final scale
    __shared__ float redP1[WAVES_PER_BLOCK], redP2[WAVES_PER_BLOCK];

    const int tid  = threadIdx.x;
    const int lane = tid & 31;
    const int wave = tid >> 5;
    const int N    = lane & 15;     // col of B/D fragment
    const int hi   = lane >> 4;     // K-half selector
    const int rowBase = blockIdx.x * ROWS_PER_BLOCK + wave * ROWS_PER_WAVE;

    // ---- Pre-phase: fill WS (row 3 zeroed), Bsum, t1 = B0.W1, t2 = (B0+B1).W2 ----
    for (int i = tid; i < 4 * WPAD; i += 128) {
        const int row = i / WPAD, col = i % WPAD;
        WS[i] = (row < 3 && col < DIM) ? W[row * DIM + col] : 0.0f;
    }
    float p1 = 0.0f, p2 = 0.0f;
    for (int d = tid; d < DIM; d += 128) {
        float b0 = B[d], b1 = B[DIM + d], b2 = B[2 * DIM + d];
        BsumS[d] = b0 + b1 + b2;
        p1 += b0 * W[DIM + d];
        p2 += (b0 + b1) * W[2 * DIM + d];
    }
    #pragma unroll
    for (int off = 16; off > 0; off >>= 1) {
        p1 += __shfl_xor(p1, off, 32);
        p2 += __shfl_xor(p2, off, 32);
    }
    if (lane == 0) { redP1[wave] = p1; redP2[wave] = p2; }
    __syncthreads();
    const float t1 = redP1[0] + redP1[1] + redP1[2] + redP1[3];
    const float t2 = redP2[0] + redP2[1] + redP2[2] + redP2[3];

    // ---- WMMA phase: D[16x16] = X[16x1024] * W^T[1024x16] (cols 0..2 used) ----
    // A fragment: lane holds x0[rowBase + N][k + 2*hi + {0,1}]
    // B fragment: lane holds W[N][k + 2*hi + {0,1}]  (zero row for N >= 3)
    const float* arow = x0 + (size_t)(rowBase + N) * DIM;
    const int wOff = (N < 3 ? N : 3) * WPAD + 2 * hi;   // zero row for N >= 3

    v8f acc[4];
    #pragma unroll
    for (int u = 0; u < 4; ++u)
        acc[u] = (v8f){0.f, 0.f, 0.f, 0.f, 0.f, 0.f, 0.f, 0.f};

    for (int kb = 0; kb < DIM; kb += 32) {
        v2f a[8], b[8];
        #pragma unroll
        for (int u = 0; u < 8; ++u) {
            const int k = kb + u * 4;
            a[u] = *(const v2f*)(arow + k + 2 * hi);       // global_load_b64
            b[u] = *(const v2f*)(&WS[wOff + k]);           // ds_load_b64 (no divergence)
        }
        #pragma unroll
        for (int u = 0; u < 8; ++u)
            acc[u & 3] = __builtin_amdgcn_wmma_f32_16x16x4_f32(
                false, a[u], false, b[u], (short)0, acc[u & 3], false, false);
    }
    v8f c = (acc[0] + acc[1]) + (acc[2] + acc[3]);

    // ---- Extract D columns 0..2: lane holds D[v + 8*hi][N] in c[v] ----
    if (N < 3) {
        #pragma unroll
        for (int v = 0; v < 8; ++v)
            dts[wave][N][v + 8 * hi] = c[v];
    }
    __syncthreads();

    // ---- Per-row scalar recurrence: alpha_{l+1} = alpha_l + alpha_l*d_l + t_l ----
    if (lane < 16) {
        const float d0 = dts[wave][0][lane];
        const float d1 = dts[wave][1][lane];
        const float d2 = dts[wave][2][lane];
        float a1 = 1.0f + d0;                // alpha0 = 1, t0 = 0
        float a2 = a1 + (a1 * d1 + t1);
        float a3 = a2 + (a2 * d2 + t2);
        alphaS[wave][lane] = a3;
    }
    __syncthreads();

    // ---- Output phase: out = alpha * x0 + Bsum (coalesced float4, x0 from L2) ----
    const float4* bsum4 = (const float4*)BsumS;
    for (int r = 0; r < ROWS_PER_WAVE; ++r) {
        const float alpha = alphaS[wave][r];
        const float4* xr = (const float4*)(x0 + (size_t)(rowBase + r) * DIM);
        float4* orow = (float4*)(out + (size_t)(rowBase + r) * DIM);
        #pragma unroll
        for (int j = lane; j < DIM / 4; j += 32) {
            float4 xv = xr[j];
            float4 bv = bsum4[j];
            float4 ov;
            ov.x = fmaf(xv.x, alpha, bv.x);
            ov.y = fmaf(xv.y, alpha, bv.y);
            ov.z = fmaf(xv.z, alpha, bv.z);
            ov.w = fmaf(xv.w, alpha, bv.w);
            orow[j] = ov;
        }
    }
}

extern "C" void kernel_launch(void* const* d_in, const int* in_sizes, int n_in,
                              void* d_out, int out_size, void* d_ws, size_t ws_size,
                              hipStream_t stream) {
    const float* x0 = (const float*)d_in[0];   // [16384*1024]
    const float* W  = (const float*)d_in[1];   // [3*1024]
    const float* B  = (const float*)d_in[2];   // [3*1024]
    float* out = (float*)d_out;

    const int batch = in_sizes[0] / DIM;       // 16384
    const int blocks = batch / ROWS_PER_BLOCK; // 256
    crossnet_wmma_kernel<<<blocks, 128, 0, stream>>>(x0, W, B, out);
}